// GCNClassifier_28355374088749
// MI455X (gfx1250) — compile-verified
//
#include <hip/hip_runtime.h>

typedef __attribute__((ext_vector_type(2))) float v2f;
typedef __attribute__((ext_vector_type(8))) float v8f;

#define D_IN  128
#define D_HID 128
#define N_CLS 2

// ---------------- degree / norm ----------------

__global__ void k_deg_init(float* __restrict__ deg, int n) {
    int i = blockIdx.x * blockDim.x + threadIdx.x;
    if (i < n) deg[i] = 1.0f;                       // self-loop contribution
}

__global__ void k_deg_count(const int* __restrict__ dst, float* __restrict__ deg, int E) {
    int e = blockIdx.x * blockDim.x + threadIdx.x;
    if (e < E) atomicAdd(&deg[dst[e]], 1.0f);
}

__global__ void k_dinv(float* __restrict__ deg, int n) {
    int i = blockIdx.x * blockDim.x + threadIdx.x;
    if (i < n) deg[i] = rsqrtf(deg[i]);             // deg >= 1 always (self-loop)
}

// ---------------- dense transform: C[n x 128] = A[n x 128] * B[128 x 128] ----------------
// WMMA f32 16x16x4, wave32. Block = 16 rows; 8 waves, wave w owns column tile w.

__global__ __launch_bounds__(256)
void k_gemm_wmma_f32(const float* __restrict__ A, const float* __restrict__ B,
                     float* __restrict__ C, int n_rows) {
    __shared__ float Alds[16 * D_IN];              // 8 KB A tile shared by 8 waves
    const int row0 = blockIdx.x * 16;

    // cooperative A-tile load, float4 granularity (512 float4s / 256 threads)
    for (int t = threadIdx.x; t < 16 * (D_IN / 4); t += blockDim.x) {
        int r  = t >> 5;                            // / (128/4)
        int c4 = t & 31;
        int gr = row0 + r; if (gr >= n_rows) gr = n_rows - 1;
        ((float4*)Alds)[t] = ((const float4*)(A + (size_t)gr * D_IN))[c4];
    }
    __syncthreads();

    const int wave = threadIdx.x >> 5;              // 0..7 -> column tile
    const int lane = threadIdx.x & 31;
    const int m    = lane & 15;                     // M index (A row / B,C column slot)
    const int half = lane >> 4;                     // K-pair selector
    const int col  = wave * 16 + m;                 // N index

    v8f c = {};
    const float* Arow = Alds + m * D_IN;

#pragma unroll
    for (int kk = 0; kk < D_IN / 4; ++kk) {
        const int k = kk * 4 + 2 * half;
        v2f a;                                      // A 16x4 fragment: VGPR0->K=k, VGPR1->K=k+1
        a.x = Arow[k];
        a.y = Arow[k + 1];
        v2f b;                                      // B 4x16 fragment (row striped across lanes)
        b.x = B[(size_t)(k)     * D_HID + col];
        b.y = B[(size_t)(k + 1) * D_HID + col];
        // 8 args: (neg_a, A, neg_b, B, c_mod, C, reuse_a, reuse_b)
        c = __builtin_amdgcn_wmma_f32_16x16x4_f32(false, a, false, b, (short)0, c, false, false);
    }

#pragma unroll
    for (int r = 0; r < 8; ++r) {                   // C layout: M = r + 8*half, N = col
        int gr = row0 + r + 8 * half;
        if (gr < n_rows) C[(size_t)gr * D_HID + col] = c[r];
    }
}

// ---------------- aggregation layer 1 (128 channels) ----------------

__global__ void k_selfloop_init(const float* __restrict__ h, const float* __restrict__ dinv,
                                float* __restrict__ agg, int n) {
    int t = blockIdx.x * blockDim.x + threadIdx.x;  // over n*128 (6.4M, fits int)
    if (t < n * D_HID) {
        int i = t >> 7;
        float di = dinv[i];
        agg[t] = di * di * h[t];
    }
}

__global__ void k_scatter_edges(const int* __restrict__ src, const int* __restrict__ dst,
                                const float* __restrict__ dinv, const float* __restrict__ h,
                                float* __restrict__ agg, int E) {
    int t = blockIdx.x * blockDim.x + threadIdx.x;  // one wave (32 lanes) per edge
    int e = t >> 5;
    int lane = t & 31;
    if (e >= E) return;
    if (e + 1 < E)                                  // prefetch next edge's source row (L2-resident)
        __builtin_prefetch(h + (size_t)src[e + 1] * D_HID + lane * 4, 0, 1);
    int s = src[e], d = dst[e];
    float norm = dinv[s] * dinv[d];
    const float4 hv = ((const float4*)(h + (size_t)s * D_HID))[lane];
    float* out = agg + (size_t)d * D_HID + lane * 4;
    atomicAdd(out + 0, norm * hv.x);
    atomicAdd(out + 1, norm * hv.y);
    atomicAdd(out + 2, norm * hv.z);
    atomicAdd(out + 3, norm * hv.w);
}

__global__ void k_bias_relu(float* __restrict__ agg, const float* __restrict__ b, int total) {
    int t = blockIdx.x * blockDim.x + threadIdx.x;
    if (t < total) {
        float v = agg[t] + b[t & (D_HID - 1)];
        agg[t] = v > 0.0f ? v : 0.0f;
    }
}

// ---------------- layer 2: dense 128->2 fused with d_out self-loop init ----------------

__global__ void k_gemm2_init_out(const float* __restrict__ h2, const float* __restrict__ W2,
                                 const float* __restrict__ b2, const float* __restrict__ dinv,
                                 float* __restrict__ z2, float* __restrict__ out, int n) {
    int i = blockIdx.x * blockDim.x + threadIdx.x;
    if (i >= n) return;
    const float4* hr = (const float4*)(h2 + (size_t)i * D_HID);
    float acc0 = 0.0f, acc1 = 0.0f;
#pragma unroll 8
    for (int c4 = 0; c4 < D_HID / 4; ++c4) {
        float4 v = hr[c4];
        int c = c4 * 4;
        acc0 += v.x * W2[(c + 0) * N_CLS + 0] + v.y * W2[(c + 1) * N_CLS + 0]
              + v.z * W2[(c + 2) * N_CLS + 0] + v.w * W2[(c + 3) * N_CLS + 0];
        acc1 += v.x * W2[(c + 0) * N_CLS + 1] + v.y * W2[(c + 1) * N_CLS + 1]
              + v.z * W2[(c + 2) * N_CLS + 1] + v.w * W2[(c + 3) * N_CLS + 1];
    }
    z2[i * N_CLS + 0] = acc0;
    z2[i * N_CLS + 1] = acc1;
    float d2 = dinv[i] * dinv[i];
    out[i * N_CLS + 0] = d2 * acc0 + b2[0];
    out[i * N_CLS + 1] = d2 * acc1 + b2[1];
}

__global__ void k_scatter2(const int* __restrict__ src, const int* __restrict__ dst,
                           const float* __restrict__ dinv, const float* __restrict__ z2,
                           float* __restrict__ out, int E) {
    int e = blockIdx.x * blockDim.x + threadIdx.x;
    if (e >= E) return;
    int s = src[e], d = dst[e];
    float norm = dinv[s] * dinv[d];
    atomicAdd(&out[d * N_CLS + 0], norm * z2[s * N_CLS + 0]);
    atomicAdd(&out[d * N_CLS + 1], norm * z2[s * N_CLS + 1]);
}

// ---------------- launcher ----------------

extern "C" void kernel_launch(void* const* d_in, const int* in_sizes, int n_in,
                              void* d_out, int out_size, void* d_ws, size_t ws_size,
                              hipStream_t stream) {
    const float* x  = (const float*)d_in[0];
    const int*   ei = (const int*)d_in[1];
    const float* W1 = (const float*)d_in[2];
    const float* b1 = (const float*)d_in[3];
    const float* W2 = (const float*)d_in[4];
    const float* b2 = (const float*)d_in[5];
    float*       out = (float*)d_out;

    const int N = in_sizes[0] / D_IN;               // 50000
    const int E = in_sizes[1] / 2;                  // 600000
    const int* src = ei;
    const int* dst = ei + E;

    float* ws = (float*)d_ws;
    size_t off = 0;
    float* dinv = ws + off; off += ((size_t)N + 255) & ~(size_t)255;
    float* h1   = ws + off; off += (size_t)N * D_HID;
    float* agg1 = ws + off; off += (size_t)N * D_HID;
    float* z2   = ws + off;                         // N * 2

    const int B = 256;
    const int total1 = N * D_HID;

    k_deg_init <<<(N + B - 1) / B, B, 0, stream>>>(dinv, N);
    k_deg_count<<<(E + B - 1) / B, B, 0, stream>>>(dst, dinv, E);
    k_dinv     <<<(N + B - 1) / B, B, 0, stream>>>(dinv, N);

    k_gemm_wmma_f32<<<(N + 15) / 16, 256, 0, stream>>>(x, W1, h1, N);

    k_selfloop_init<<<(total1 + B - 1) / B, B, 0, stream>>>(h1, dinv, agg1, N);
    {
        long t = (long)E * 32;
        k_scatter_edges<<<(unsigned)((t + B - 1) / B), B, 0, stream>>>(src, dst, dinv, h1, agg1, E);
    }
    k_bias_relu<<<(total1 + B - 1) / B, B, 0, stream>>>(agg1, b1, total1);

    k_gemm2_init_out<<<(N + B - 1) / B, B, 0, stream>>>(agg1, W2, b2, dinv, z2, out, N);
    k_scatter2      <<<(E + B - 1) / B, B, 0, stream>>>(src, dst, dinv, z2, out, E);
}